// Laplacian2D_15229954031889
// MI455X (gfx1250) — compile-verified
//
#include <hip/hip_runtime.h>
#include <cstdint>

// Problem constants (from reference): u is (B,C,H,W) f32, D is (ND,C) f32.
#define ND       3
#define BATCH    4
#define CHANNELS 64
#define HEIGHT   512
#define WIDTH    512

#define HALO   16                       // max dilation
#define RPITCH (WIDTH + 2 * HALO)       // 544 floats; row pitch 2176B (16B aligned)

#define TX 32                           // one wave == one full row
#define ROWS_PER_BLOCK 8                // block = 32x8 = 256 threads = 8 wave32
#define NGROUPS 4                       // 4 strided v4f groups per thread
#define GSTRIDE 128                     // columns between a thread's groups (512B)

typedef __attribute__((ext_vector_type(4))) float v4f;
typedef __attribute__((ext_vector_type(2))) float v2f;   // -> v_pk_*_f32 if legal

// gfx1250 async global->LDS path (ASYNCcnt) — confirmed emitting
// global_load_async_to_lds_b128 (s_clause'd) on this toolchain.
#if defined(__HIP_DEVICE_COMPILE__) && __has_builtin(__builtin_amdgcn_global_load_async_to_lds_b128)
#define USE_ASYNC_LDS 1
#else
#define USE_ASYNC_LDS 0
#endif

#if USE_ASYNC_LDS
typedef __attribute__((ext_vector_type(4))) int v4i;
typedef __attribute__((address_space(1))) v4i gv4i;   // global AS pointee
typedef __attribute__((address_space(3))) v4i lv4i;   // LDS AS pointee
#endif

__device__ __forceinline__ v4f ld4(const float* p) {
  return *reinterpret_cast<const v4f*>(p);
}

__device__ __forceinline__ float softplus_f(float x) {
  return fmaxf(x, 0.0f) + log1pf(expf(-fabsf(x)));   // stable, matches jax.nn.softplus
}

__global__ __launch_bounds__(TX * ROWS_PER_BLOCK)
void lap_multi_dilation_kernel(const float* __restrict__ u,
                               const float* __restrict__ D,
                               float* __restrict__ out) {
  const int tx = threadIdx.x;
  const int ty = threadIdx.y;
  const int cw = tx * 4;                              // base column of group 0
  const int h  = blockIdx.y * ROWS_PER_BLOCK + ty;
  const int p  = blockIdx.z;                          // plane = b*C + c
  const int c  = p & (CHANNELS - 1);

  const size_t planeOff = (size_t)p * (size_t)(HEIGHT * WIDTH);
  const float* P    = u + planeOff;
  const float* rowp = P + (uint32_t)h * WIDTH;

  __shared__ float rowbuf[ROWS_PER_BLOCK][RPITCH];    // halo'd center rows
  __shared__ float ssc[ND];

  // ---- stage center row into LDS: 4 async b128, IOFFSET covers both sides ----
#if USE_ASYNC_LDS
  {
    gv4i* gsrc = (gv4i*)(rowp + cw);
    lv4i* ldst = (lv4i*)&rowbuf[ty][HALO + cw];
    __builtin_amdgcn_global_load_async_to_lds_b128(gsrc, ldst, 0,    0);
    __builtin_amdgcn_global_load_async_to_lds_b128(gsrc, ldst, 512,  0);  // +128 cols
    __builtin_amdgcn_global_load_async_to_lds_b128(gsrc, ldst, 1024, 0);  // +256 cols
    __builtin_amdgcn_global_load_async_to_lds_b128(gsrc, ldst, 1536, 0);  // +384 cols
  }
#else
#pragma unroll
  for (int j = 0; j < NGROUPS; ++j)
    *reinterpret_cast<v4f*>(&rowbuf[ty][HALO + cw + j * GSTRIDE]) =
        ld4(rowp + cw + j * GSTRIDE);
#endif

  // ---- halo fill: replicate edge element (edge-pad == index clamp) ----
  if (tx == 0) {
    const float e = rowp[0];
    const v4f e4 = {e, e, e, e};
    v4f* hb = reinterpret_cast<v4f*>(&rowbuf[ty][0]);
    hb[0] = e4; hb[1] = e4; hb[2] = e4; hb[3] = e4;
  }
  if (tx == TX - 1) {
    const float e = rowp[WIDTH - 1];
    const v4f e4 = {e, e, e, e};
    v4f* hb = reinterpret_cast<v4f*>(&rowbuf[ty][HALO + WIDTH]);
    hb[0] = e4; hb[1] = e4; hb[2] = e4; hb[3] = e4;
  }

  // cross-block L2 prefetch
  {
    int hp = h + 64; if (hp > HEIGHT - 1) hp = HEIGHT - 1;
    __builtin_prefetch(P + (uint32_t)hp * WIDTH + cw, 0, 1);
  }

  // ---- per-channel softplus scales (block-uniform channel) ----
  if (ty == 0 && tx < ND) ssc[tx] = softplus_f(D[tx * CHANNELS + c]);

  // ---- vertical tap row bases (32-bit offsets -> saddr-friendly) ----
  int hu1  = h - 1;  if (hu1  < 0) hu1  = 0;
  int hd1  = h + 1;  if (hd1  > HEIGHT - 1) hd1  = HEIGHT - 1;
  int hu4  = h - 4;  if (hu4  < 0) hu4  = 0;
  int hd4  = h + 4;  if (hd4  > HEIGHT - 1) hd4  = HEIGHT - 1;
  int hu16 = h - 16; if (hu16 < 0) hu16 = 0;
  int hd16 = h + 16; if (hd16 > HEIGHT - 1) hd16 = HEIGHT - 1;

  const float* pu1  = P + ((uint32_t)hu1  * WIDTH + cw);
  const float* pd1  = P + ((uint32_t)hd1  * WIDTH + cw);
  const float* pu4  = P + ((uint32_t)hu4  * WIDTH + cw);
  const float* pd4  = P + ((uint32_t)hd4  * WIDTH + cw);
  const float* pu16 = P + ((uint32_t)hu16 * WIDTH + cw);
  const float* pd16 = P + ((uint32_t)hd16 * WIDTH + cw);

#if USE_ASYNC_LDS
#if __has_builtin(__builtin_amdgcn_s_wait_asynccnt)
  __builtin_amdgcn_s_wait_asynccnt(0);
#else
  asm volatile("s_wait_asynccnt 0" ::: "memory");
#endif
#endif
  __syncthreads();

  const float s1  = ssc[0];
  const float s4  = ssc[1];
  const float s16 = ssc[2];
  const float m4  = -4.0f * (s1 + s4 + s16);

  // lbp[go + k] == column (group_base + k - 16); all LDS taps are imm offsets.
  const float* lbp = &rowbuf[ty][cw];
  float* op = out + planeOff + (uint32_t)h * WIDTH + cw;

#pragma unroll
  for (int j = 0; j < NGROUPS; ++j) {
    const int go = j * GSTRIDE;

    // vertical taps: 6 x global_load_b128, imm offset j*512B
    const v4f uv1  = ld4(pu1  + go);
    const v4f dv1  = ld4(pd1  + go);
    const v4f uv4  = ld4(pu4  + go);
    const v4f dv4  = ld4(pd4  + go);
    const v4f uv16 = ld4(pu16 + go);
    const v4f dv16 = ld4(pd16 + go);

    // horizontal window: 5 x ds_load_b128, imm offsets, conflict-free (16B lane stride)
    const v4f l16 = ld4(lbp + go);        // cols base-16 .. base-13
    const v4f l4  = ld4(lbp + go + 12);   // cols base-4  .. base-1
    const v4f cc  = ld4(lbp + go + 16);   // cols base    .. base+3
    const v4f r4  = ld4(lbp + go + 20);   // cols base+4  .. base+7
    const v4f r16 = ld4(lbp + go + 32);   // cols base+16 .. base+19

    // d=1 shifted windows, in packed halves.
    // l1 = (l4.w, cc.x, cc.y, cc.z), r1 = (cc.y, cc.z, cc.w, r4.x)
    // -> l1.hi == r1.lo == cc.yz (shared swizzle)
    const v2f l1lo = {l4.w, cc.x};
    const v2f mid  = cc.yz;
    const v2f r1hi = {cc.w, r4.x};

    // packed-f32 accumulation (v_pk_add_f32 / v_pk_fma_f32 if legal)
    const v2f alo = m4 * cc.lo
                  + s1  * ((uv1.lo  + dv1.lo)  + (l1lo   + mid))
                  + s4  * ((uv4.lo  + dv4.lo)  + (l4.lo  + r4.lo))
                  + s16 * ((uv16.lo + dv16.lo) + (l16.lo + r16.lo));
    const v2f ahi = m4 * cc.hi
                  + s1  * ((uv1.hi  + dv1.hi)  + (mid    + r1hi))
                  + s4  * ((uv4.hi  + dv4.hi)  + (l4.hi  + r4.hi))
                  + s16 * ((uv16.hi + dv16.hi) + (l16.hi + r16.hi));

    const v4f acc = {alo.x, alo.y, ahi.x, ahi.y};
    __builtin_nontemporal_store(acc, reinterpret_cast<v4f*>(op + go));
  }
}

extern "C" void kernel_launch(void* const* d_in, const int* in_sizes, int n_in,
                              void* d_out, int out_size, void* d_ws, size_t ws_size,
                              hipStream_t stream) {
  (void)in_sizes; (void)n_in; (void)d_ws; (void)ws_size; (void)out_size;

  const float* u = (const float*)d_in[0];   // (4,64,512,512) f32
  const float* D = (const float*)d_in[1];   // (3,64) f32
  float* out = (float*)d_out;               // (4,64,512,512) f32

  dim3 block(TX, ROWS_PER_BLOCK, 1);                        // 256 threads = 8 wave32
  dim3 grid(1, HEIGHT / ROWS_PER_BLOCK, BATCH * CHANNELS);  // (1,64,256)
  hipLaunchKernelGGL(lap_multi_dilation_kernel, grid, block, 0, stream, u, D, out);
}